// Spectral_DCM_Model_20469814133343
// MI455X (gfx1250) — compile-verified
//
#include <hip/hip_runtime.h>

// Spectral DCM model, analytically reduced.
//   X = i*w*I - A*J  (A scalar, J = ones(R,R))  =>  Sherman-Morrison:
//   X^-1 = p*I + s*J,  p = 1/(i w),  s = A / (i w (i w - A R))
//   res  = gv*|hrf|^2 * (|p|^2 * I + (2 Re(p conj(s)) + R |s|^2) * J) + ge*I
// => per frequency only TWO real scalars: diag value and off-diag value.
// Output: (R*R, F) complex64, row-major, interleaved (re, im); imag == 0.

#define RNUM 64
#define FNUM 4096

__device__ __forceinline__ float2 cmul(float2 a, float2 b) {
    return make_float2(a.x * b.x - a.y * b.y, a.x * b.y + a.y * b.x);
}

// ---------------------------------------------------------------------------
// Kernel 1: per-frequency scalar terms -> d_ws as float2 {diag, offdiag}
// ---------------------------------------------------------------------------
__global__ void __launch_bounds__(256) freq_terms_kernel(
    const float* __restrict__ freqs,
    const float* __restrict__ alphas_e, const float* __restrict__ betas_e,
    const float* __restrict__ alphas_v, const float* __restrict__ betas_v,
    const float* __restrict__ Aptr,
    float2* __restrict__ terms) {
    int f = blockIdx.x * blockDim.x + threadIdx.x;
    if (f >= FNUM) return;

    const float w = freqs[f];
    const float A = Aptr[0];

    // hrf = (6*(1+iw)^10 - 1) / (5*(1+iw)^16); only |hrf|^2 is needed.
    float2 z   = make_float2(1.0f, w);
    float2 z2  = cmul(z, z);
    float2 z4  = cmul(z2, z2);
    float2 z8  = cmul(z4, z4);
    float2 z10 = cmul(z8, z2);
    float2 z16 = cmul(z8, z8);
    float2 num = make_float2(6.0f * z10.x - 1.0f, 6.0f * z10.y);
    float2 den = make_float2(5.0f * z16.x, 5.0f * z16.y);
    const float h2 = (num.x * num.x + num.y * num.y) /
                     (den.x * den.x + den.y * den.y);   // |hrf|^2

    const float gv = alphas_v[0] * powf(w, -betas_v[0]);
    const float ge = alphas_e[0] * powf(w, -betas_e[0]);

    const float Rf = (float)RNUM;
    const float p2 = 1.0f / (w * w);                    // |p|^2, p = -i/w
    // s = A / D,  D = (iw)^2 - iw*A*R = (-w^2) + i*(-A*R*w)
    const float Dr  = -w * w;
    const float Di  = -A * Rf * w;
    const float Dm2 = Dr * Dr + Di * Di;
    const float sr  =  A * Dr / Dm2;
    const float si  = -A * Di / Dm2;
    // 2*Re(p*conj(s)) = -2*si/w  (real);  plus R*|s|^2 (real)
    const float off_factor = -2.0f * si / w + Rf * (sr * sr + si * si);

    const float K    = gv * h2;
    const float offd = K * off_factor;
    const float diag = K * p2 + offd + ge;

    terms[f] = make_float2(diag, offd);
}

// ---------------------------------------------------------------------------
// Kernel 2: stream 128 MiB of output. One block per output row (reg index).
// Row layout in floats: [v(f0), 0, v(f1), 0, ...]  (32 KB per row).
// Row is staged in LDS, then drained via the CDNA5 async LDS->global engine
// (global_store_async_from_lds_b128, tracked by ASYNCcnt).
// ---------------------------------------------------------------------------
__global__ void __launch_bounds__(256) write_rows_kernel(
    const float4* __restrict__ terms4,   // pairs of float2 terms, f0/f0+1
    float* __restrict__ out) {
    __shared__ float4 smem[2048];        // 32 KB = one full output row

    const int t   = threadIdx.x;
    const int reg = blockIdx.x;                    // 0 .. R*R-1
    const bool isDiag = (reg % (RNUM + 1)) == 0;   // reg == r*65 <=> r==c

    // Build the row in LDS: float4 i covers frequencies f=2i and f=2i+1.
    #pragma unroll
    for (int k = 0; k < 8; ++k) {
        const int i = t + k * 256;                 // 0 .. 2047
        const float4 ab = terms4[i];               // {diag0,off0,diag1,off1}
        const float v0 = isDiag ? ab.x : ab.y;
        const float v1 = isDiag ? ab.z : ab.w;
        smem[i] = make_float4(v0, 0.0f, v1, 0.0f);
    }
    __syncthreads();

    // Drain LDS -> global with async b128 stores (512 B per wave-instruction).
    const unsigned lds_base = (unsigned)(size_t)(&smem[0]);
    const unsigned long long gbase =
        (unsigned long long)(size_t)out + (unsigned long long)reg * 32768ull;

    #pragma unroll
    for (int k = 0; k < 8; ++k) {
        const int i = t + k * 256;
        const unsigned long long ga  = gbase + (unsigned long long)(i * 16);
        const unsigned           lof = lds_base + (unsigned)(i * 16);
        asm volatile("global_store_async_from_lds_b128 %0, %1, off"
                     :: "v"(ga), "v"(lof) : "memory");
    }
    // Ensure all async LDS->global traffic retired before LDS is released.
    asm volatile("s_wait_asynccnt 0" ::: "memory");
}

// ---------------------------------------------------------------------------
extern "C" void kernel_launch(void* const* d_in, const int* in_sizes, int n_in,
                              void* d_out, int out_size, void* d_ws, size_t ws_size,
                              hipStream_t stream) {
    const float* freqs    = (const float*)d_in[0];
    const float* alphas_e = (const float*)d_in[1];
    const float* betas_e  = (const float*)d_in[2];
    const float* alphas_v = (const float*)d_in[3];
    const float* betas_v  = (const float*)d_in[4];
    const float* Aptr     = (const float*)d_in[5];

    float2* terms = (float2*)d_ws;       // 4096 * 8 B = 32 KB scratch
    float*  out   = (float*)d_out;       // (R*R, F) complex64, interleaved

    freq_terms_kernel<<<FNUM / 256, 256, 0, stream>>>(
        freqs, alphas_e, betas_e, alphas_v, betas_v, Aptr, terms);

    write_rows_kernel<<<RNUM * RNUM, 256, 0, stream>>>(
        (const float4*)terms, out);
}